// CapsNet_84224308675202
// MI455X (gfx1250) — compile-verified
//
#include <hip/hip_runtime.h>
#include <math.h>
#include <stdint.h>

typedef __attribute__((ext_vector_type(2))) float v2f;
typedef __attribute__((ext_vector_type(8))) float v8f;

#define B_SZ 256
#define I_SZ 1152
#define O_SZ 10
#define DIN  8
#define DOUT 16
#define KTOT (I_SZ * DIN)    // 9216 flattened (i,d)
#define NTOT (O_SZ * DOUT)   // 160  flattened (o,e)
#define ROUTING_ITERS 3
#define KSPLIT 8             // K-dim split for s-GEMM
#define KCHUNK (KTOT / KSPLIT)   // 1152 floats of K per block
#define XPAD 4               // LDS row padding (floats) to dodge bank conflicts

// ---- gfx1250 async global->LDS copy (ASYNCcnt-tracked, bypasses VGPRs) ----
__device__ __forceinline__ void async_copy_b128(uint32_t lds_byte_addr,
                                                const float* gsrc) {
    asm volatile("global_load_async_to_lds_b128 %0, %1, off"
                 :: "v"(lds_byte_addr), "v"(gsrc)
                 : "memory");
}
__device__ __forceinline__ void wait_async0() {
    asm volatile("s_wait_asynccnt 0x0" ::: "memory");
}
__device__ __forceinline__ uint32_t lds_addr_of(const void* p) {
    return (uint32_t)(uintptr_t)p;   // low 32 bits of generic ptr = LDS offset
}

// ---------------------------------------------------------------- init logits
__global__ void init_b_kernel(float* __restrict__ blog) {
    int t = blockIdx.x * blockDim.x + threadIdx.x;
    if (t < I_SZ * O_SZ) blog[t] = 0.0f;
}

// ------------------------------------------------- softmax over out-capsules
__global__ void softmax_kernel(const float* __restrict__ blog,
                               float* __restrict__ c) {
    int i = blockIdx.x * blockDim.x + threadIdx.x;
    if (i >= I_SZ) return;
    const float* br = blog + i * O_SZ;
    float m = br[0];
    #pragma unroll
    for (int o = 1; o < O_SZ; ++o) m = fmaxf(m, br[o]);
    float e[O_SZ];
    float sum = 0.0f;
    #pragma unroll
    for (int o = 0; o < O_SZ; ++o) { e[o] = __expf(br[o] - m); sum += e[o]; }
    float inv = 1.0f / sum;
    #pragma unroll
    for (int o = 0; o < O_SZ; ++o) c[i * O_SZ + o] = e[o] * inv;
}

// --------------------------------------------------------- s partials (WMMA)
// s = x2 @ (c∘W);  x2 = x as [B, 9216] row-major.
// Grid (16 batch-tiles, 10 capsules, 8 K-splits), block = 128 = 4 waves.
// x-slab [16 x 1152] is staged to LDS with async b128 copies (coalesced),
// W stays in global (already lane-coalesced, L2-resident).
__global__ __launch_bounds__(128) void s_gemm_kernel(const float* __restrict__ x,
                                                     const float* __restrict__ W,
                                                     const float* __restrict__ c,
                                                     float* __restrict__ s_part) {
    const int mt   = blockIdx.x;            // batch tile 0..15
    const int o    = blockIdx.y;            // out capsule 0..9
    const int kc   = blockIdx.z;            // K split 0..7
    const int lane = threadIdx.x & 31;
    const int wave = threadIdx.x >> 5;
    const int mrow  = lane & 15;
    const int col   = lane & 15;
    const int khalf = (lane >> 4) * 2;

    __shared__ __align__(16) float xs[16][KCHUNK + XPAD];
    __shared__ float red[4][32][8];

    // ---- stage x[mt*16 .. +15][kc*1152 .. +1152) into LDS, coalesced ----
    {
        const int nB128 = 16 * KCHUNK / 4;          // 4608 16-byte transfers
        for (int q = threadIdx.x; q < nB128; q += 128) {
            int r    = q / (KCHUNK / 4);
            int col4 = (q % (KCHUNK / 4)) * 4;
            const float* gsrc = x + (size_t)(mt * 16 + r) * KTOT
                                  + (size_t)kc * KCHUNK + col4;
            async_copy_b128(lds_addr_of(&xs[r][col4]), gsrc);
        }
        wait_async0();
        __syncthreads();
    }

    // ---- WMMA main loop: each wave covers 288 K-values (72 steps) ----
    v8f acc = {};
    const int kk0 = wave * (KCHUNK / 4);            // 288 per wave
    const int kk1 = kk0 + (KCHUNK / 4);
    for (int kk = kk0; kk < kk1; kk += 4) {
        v2f a;                                       // A frag from LDS
        a.x = xs[mrow][kk + khalf];
        a.y = xs[mrow][kk + khalf + 1];
        const int kg = kc * KCHUNK + kk;             // global K index
        const int i  = kg >> 3;
        const int d  = (kg & 7) + khalf;
        const float scale = c[i * O_SZ + o];
        const float* wr = W + (((size_t)(i * O_SZ + o) * DIN + d) * DOUT) + col;
        v2f b;
        b.x = wr[0]    * scale;
        b.y = wr[DOUT] * scale;
        acc = __builtin_amdgcn_wmma_f32_16x16x4_f32(
            false, a, false, b, (short)0, acc, false, false);
    }

    // ---- reduce 4 waves, write this K-split's partial tile ----
    #pragma unroll
    for (int r = 0; r < 8; ++r) red[wave][lane][r] = acc[r];
    __syncthreads();
    if (wave == 0) {
        float* sp = s_part + (size_t)kc * B_SZ * NTOT;
        #pragma unroll
        for (int r = 0; r < 8; ++r) {
            float t = red[0][lane][r] + red[1][lane][r] +
                      red[2][lane][r] + red[3][lane][r];
            int m = mt * 16 + r + ((lane >> 4) * 8);
            sp[(size_t)m * NTOT + o * DOUT + col] = t;
        }
    }
}

// ----------------------------------------------- sum the K-split partials
__global__ void s_reduce_kernel(const float* __restrict__ s_part,
                                float* __restrict__ s) {
    int t = blockIdx.x * blockDim.x + threadIdx.x;
    if (t >= B_SZ * NTOT) return;
    float acc = 0.0f;
    #pragma unroll
    for (int k = 0; k < KSPLIT; ++k)
        acc += s_part[(size_t)k * B_SZ * NTOT + t];
    s[t] = acc;
}

// ------------------------------------------------------------------- squash
__global__ void squash_kernel(const float* __restrict__ s,
                              float* __restrict__ v,
                              float* __restrict__ out, int write_out) {
    int t = blockIdx.x * blockDim.x + threadIdx.x;   // (b*O + o)
    if (t >= B_SZ * O_SZ) return;
    const float* sr = s + (size_t)t * DOUT;
    float ss = 0.0f;
    #pragma unroll
    for (int e = 0; e < DOUT; ++e) { float q = sr[e]; ss += q * q; }
    float scale = ss / ((1.0f + ss) * sqrtf(ss));
    float* vr = v + (size_t)t * DOUT;
    #pragma unroll
    for (int e = 0; e < DOUT; ++e) {
        float q = sr[e] * scale;
        vr[e] = q;
        if (write_out) out[(size_t)t * DOUT + e] = q;
    }
}

// --------------------------------------------------------------- G = x2^T @ v
// A' rows = flattened (i,d): A'[m'][k'] = x[k'*9216 + m']  (lane-coalesced).
__global__ __launch_bounds__(32) void g_gemm_kernel(const float* __restrict__ x,
                                                    const float* __restrict__ v,
                                                    float* __restrict__ G) {
    const int mt   = blockIdx.x;           // (i,d) tile 0..575
    const int o    = blockIdx.y;
    const int lane = threadIdx.x;
    const int mrow  = lane & 15;
    const int col   = lane & 15;
    const int khalf = (lane >> 4) * 2;

    v8f acc = {};
    const int mbase = mt * 16 + mrow;
    for (int k = 0; k < B_SZ; k += 4) {
        v2f a;
        a.x = x[(size_t)(k + khalf)     * KTOT + mbase];
        a.y = x[(size_t)(k + khalf + 1) * KTOT + mbase];
        const float* vr = v + (size_t)(k + khalf) * NTOT + o * DOUT + col;
        v2f b;
        b.x = vr[0];
        b.y = vr[NTOT];
        acc = __builtin_amdgcn_wmma_f32_16x16x4_f32(
            false, a, false, b, (short)0, acc, false, false);
    }
    #pragma unroll
    for (int r = 0; r < 8; ++r) {
        int m = mt * 16 + r + ((lane >> 4) * 8);
        G[(size_t)m * NTOT + o * DOUT + col] = acc[r];
    }
}

// --------------------------------- b += mean_b agreement, via W·G contraction
__global__ void agree_kernel(const float* __restrict__ W,
                             const float* __restrict__ G,
                             float* __restrict__ blog) {
    int t = blockIdx.x * blockDim.x + threadIdx.x;   // i*O + o
    if (t >= I_SZ * O_SZ) return;
    int i = t / O_SZ, o = t % O_SZ;
    const float* wr = W + (size_t)t * DIN * DOUT;
    float sum = 0.0f;
    #pragma unroll
    for (int d = 0; d < DIN; ++d) {
        const float* gr = G + (size_t)(i * DIN + d) * NTOT + o * DOUT;
        #pragma unroll
        for (int e = 0; e < DOUT; ++e) sum += wr[d * DOUT + e] * gr[e];
    }
    blog[t] += sum * (1.0f / B_SZ);
}

// ------------------------------------------------------------------- launch
extern "C" void kernel_launch(void* const* d_in, const int* in_sizes, int n_in,
                              void* d_out, int out_size, void* d_ws, size_t ws_size,
                              hipStream_t stream) {
    const float* x = (const float*)d_in[0];
    const float* W = (const float*)d_in[1];
    float* out = (float*)d_out;

    float* ws     = (float*)d_ws;
    float* blog   = ws;                          // 11520
    float* c      = blog   + I_SZ * O_SZ;        // 11520
    float* s      = c      + I_SZ * O_SZ;        // 40960
    float* v      = s      + B_SZ * NTOT;        // 40960
    float* G      = v      + B_SZ * NTOT;        // 1474560
    float* s_part = G      + (size_t)KTOT * NTOT; // 8*40960 (~7.6 MB total)

    init_b_kernel<<<(I_SZ * O_SZ + 255) / 256, 256, 0, stream>>>(blog);

    for (int it = 0; it < ROUTING_ITERS; ++it) {
        softmax_kernel<<<(I_SZ + 127) / 128, 128, 0, stream>>>(blog, c);
        s_gemm_kernel<<<dim3(16, O_SZ, KSPLIT), 128, 0, stream>>>(x, W, c, s_part);
        s_reduce_kernel<<<(B_SZ * NTOT + 255) / 256, 256, 0, stream>>>(s_part, s);
        const int last = (it == ROUTING_ITERS - 1);
        squash_kernel<<<(B_SZ * O_SZ + 127) / 128, 128, 0, stream>>>(s, v, out, last);
        if (!last) {
            g_gemm_kernel<<<dim3(KTOT / 16, O_SZ), 32, 0, stream>>>(x, v, G);
            agree_kernel<<<(I_SZ * O_SZ + 127) / 128, 128, 0, stream>>>(W, G, blog);
        }
    }
}